// LatticeLSTM_16355235463808
// MI455X (gfx1250) — compile-verified
//
#include <hip/hip_runtime.h>

// ---------------------------------------------------------------------------
// LatticeLSTM on MI455X (gfx1250, wave32)
//  Phase 1: time-parallel projections as fp32 WMMA GEMMs (V_WMMA_F32_16X16X4_F32)
//  Phase 2: single-WGP persistent sequential recurrence (LDS-resident state)
// ---------------------------------------------------------------------------

typedef __attribute__((ext_vector_type(2))) float v2f;
typedef __attribute__((ext_vector_type(8))) float v8f;

#define T_LEN  512
#define D_IN   256
#define HDIM   512
#define G3H    1536      // 3*HDIM
#define EDIM   256
#define KCON   2
#define KPRD   2
#define NMATCH 340

// ---------------------------------------------------------------------------
// fp32 WMMA GEMM: C(M,N) = A(M,K) @ B(K,N), row-major, optional A-row gather.
// One wave32 per 16x16 C tile. M,N multiples of 16; K multiple of 4.
// Fragment layouts per CDNA5 ISA 7.12.2 (32-bit A 16x4 / 32-bit C 16x16).
// ---------------------------------------------------------------------------
__global__ __launch_bounds__(256) void wmma_gemm_f32(
    const float* __restrict__ A, const float* __restrict__ B,
    float* __restrict__ C, int M, int N, int K,
    const int* __restrict__ rowidx)
{
  const int lane   = threadIdx.x & 31;
  const int wave   = threadIdx.x >> 5;
  const int tilesN = N >> 4;
  const int nTiles = (M >> 4) * tilesN;
  const int tile   = blockIdx.x * 8 + wave;
  if (tile >= nTiles) return;                 // whole-wave exit: EXEC stays all-1s
  const int tm = tile / tilesN;
  const int tn = tile % tilesN;

  // A: lanes 0-15 hold M=lane with K=kb+0/kb+1; lanes 16-31 hold K=kb+2/kb+3
  const int  mrow  = (tm << 4) + (lane & 15);
  const long arow  = rowidx ? (long)rowidx[mrow] : (long)mrow;
  const int  ncol  = (tn << 4) + (lane & 15);
  const int  khalf = (lane >> 4) << 1;        // 0 (lanes 0-15) or 2 (lanes 16-31)

  v8f acc = {};
  for (int kb = 0; kb < K; kb += 4) {
    v2f a, b;
    a.x = A[arow * (long)K + kb + khalf + 0];
    a.y = A[arow * (long)K + kb + khalf + 1];
    b.x = B[(long)(kb + khalf + 0) * N + ncol];
    b.y = B[(long)(kb + khalf + 1) * N + ncol];
    // (neg_a, A, neg_b, B, c_mod, C, reuse_a, reuse_b)
    acc = __builtin_amdgcn_wmma_f32_16x16x4_f32(
        false, a, false, b, (short)0, acc, false, false);
  }

  // C 16x16 f32: VGPR r -> row M=r (lanes 0-15) / M=r+8 (lanes 16-31), N=lane%16
  const int crow0 = (tm << 4) + ((lane >> 4) << 3);
#pragma unroll
  for (int r = 0; r < 8; ++r)
    C[(long)(crow0 + r) * N + ncol] = acc[r];
}

__device__ __forceinline__ float sigf(float x) { return 1.0f / (1.0f + __expf(-x)); }

// ---------------------------------------------------------------------------
// Sequential lattice recurrence. One block of 1024 threads (32 wave32s / WGP).
// h, c, gates, consumed skip-cells live in LDS. Skip-cell store in global ws.
// ---------------------------------------------------------------------------
__global__ __launch_bounds__(1024) void lattice_recurrence(
    const float* __restrict__ Gx,    // (T, 3H)  x@Wih
    const float* __restrict__ Ax,    // (T, H)   x@aWih
    const float* __restrict__ Wg,    // (T*KP, 3H) emb[prod_word]@wWih
    const float* __restrict__ Whh,   // (H, 3H)
    const float* __restrict__ bias,  // (3H)
    const float* __restrict__ aWhh,  // (H, H)
    const float* __restrict__ ab,    // (H)
    const float* __restrict__ wWhh,  // (H, 3H)
    const float* __restrict__ wb,    // (3H)
    const int*   __restrict__ prod_dest,  // (T, KP)
    const int*   __restrict__ prod_mask,  // (T, KP)
    const int*   __restrict__ cons_idx,   // (T, KC)
    const int*   __restrict__ cons_mask,  // (T, KC)
    float*       __restrict__ storeG,     // (NMATCH+1, H), pre-zeroed
    float*       __restrict__ out)        // hs (T,H) then cs (T,H)
{
  __shared__ float h[HDIM], c[HDIM];
  __shared__ float cin0[HDIM], cin1[HDIM];
  __shared__ float sg[G3H];
  __shared__ float ra0[HDIM], ra1[HDIM];
  __shared__ float rw[G3H];

  const int tid = threadIdx.x;
  if (tid < HDIM) { h[tid] = 0.0f; c[tid] = 0.0f; }
  __syncthreads();

  for (int t = 0; t < T_LEN; ++t) {
    const int   ci0 = cons_idx[t * KCON + 0];
    const int   ci1 = cons_idx[t * KCON + 1];
    const float cm0 = (float)cons_mask[t * KCON + 0];
    const float cm1 = (float)cons_mask[t * KCON + 1];

    // ---- 1. gates = Gx[t] + b + h @ Whh ; load consumed skip cells ----
    {
      const int j0 = tid, j1 = tid + 1024;
      float acc0 = 0.0f, acc1 = 0.0f;
      for (int k = 0; k < HDIM; ++k) {
        if ((k & 127) == 0 && k + 128 < HDIM)
          __builtin_prefetch(&Whh[(long)(k + 128) * G3H + j0], 0, 0);
        const float hk = h[k];
        acc0 = fmaf(hk, Whh[(long)k * G3H + j0], acc0);
        if (j1 < G3H) acc1 = fmaf(hk, Whh[(long)k * G3H + j1], acc1);
      }
      sg[j0] = Gx[(long)t * G3H + j0] + bias[j0] + acc0;
      if (j1 < G3H) sg[j1] = Gx[(long)t * G3H + j1] + bias[j1] + acc1;
    }
    if (tid < HDIM) cin0[tid]        = storeG[(long)ci0 * HDIM + tid];
    else            cin1[tid - HDIM] = storeG[(long)ci1 * HDIM + (tid - HDIM)];
    __syncthreads();

    // ---- 2. recurrent part of alpha: ra_r = c_in[r] @ aWhh ----
    {
      const int r = tid >> 9, j = tid & (HDIM - 1);
      const float* cin = r ? cin1 : cin0;
      float acc = 0.0f;
      for (int k = 0; k < HDIM; ++k)
        acc = fmaf(cin[k], aWhh[(long)k * HDIM + j], acc);
      (r ? ra1 : ra0)[j] = acc;
    }
    __syncthreads();

    // ---- 3. element-wise cell update ----
    if (tid < HDIM) {
      const int j = tid;
      const float ig = sigf(sg[j]);
      const float og = sigf(sg[HDIM + j]);
      const float gg = tanhf(sg[2 * HDIM + j]);
      const float ax = Ax[(long)t * HDIM + j] + ab[j];
      const float a0 = __expf(sigf(ax + ra0[j])) * cm0;
      const float a1 = __expf(sigf(ax + ra1[j])) * cm1;
      const float ei = __expf(ig);
      const float c_skip  = (gg * ei + cin0[j] * a0 + cin1[j] * a1) / (ei + a0 + a1);
      const float c_plain = (1.0f - ig) * c[j] + ig * gg;
      const float c1 = (cm0 + cm1 > 0.0f) ? c_skip : c_plain;
      const float h1 = og * tanhf(c1);
      c[j] = c1;
      h[j] = h1;
      out[(long)t * HDIM + j] = h1;                              // hs
      out[(long)T_LEN * HDIM + (long)t * HDIM + j] = c1;         // cs
    }
    __syncthreads();

    // ---- 4. word-LSTM recurrent part: rw = h1 @ wWhh (shared by both words) ----
    {
      const int j0 = tid, j1 = tid + 1024;
      float acc0 = 0.0f, acc1 = 0.0f;
      for (int k = 0; k < HDIM; ++k) {
        if ((k & 127) == 0 && k + 128 < HDIM)
          __builtin_prefetch(&wWhh[(long)(k + 128) * G3H + j0], 0, 0);
        const float hk = h[k];
        acc0 = fmaf(hk, wWhh[(long)k * G3H + j0], acc0);
        if (j1 < G3H) acc1 = fmaf(hk, wWhh[(long)k * G3H + j1], acc1);
      }
      rw[j0] = acc0;
      if (j1 < G3H) rw[j1] = acc1;
    }
    __syncthreads();

    // ---- 5. word cells -> skip-cell store scatter ----
    {
      const int p = tid >> 9, j = tid & (HDIM - 1);
      const float* wrow = Wg + (long)(t * KPRD + p) * G3H;
      const float f2 = wrow[j]            + wb[j]            + rw[j];
      const float i2 = wrow[HDIM + j]     + wb[HDIM + j]     + rw[HDIM + j];
      const float g2 = wrow[2 * HDIM + j] + wb[2 * HDIM + j] + rw[2 * HDIM + j];
      const float ct = sigf(f2) * c[j] + sigf(i2) * tanhf(g2);
      const int dest = prod_mask[t * KPRD + p] ? prod_dest[t * KPRD + p] : NMATCH;
      storeG[(long)dest * HDIM + j] = ct;   // masked rows -> dummy slot (never read)
    }
    __syncthreads();
  }
}

// ---------------------------------------------------------------------------
extern "C" void kernel_launch(void* const* d_in, const int* in_sizes, int n_in,
                              void* d_out, int out_size, void* d_ws, size_t ws_size,
                              hipStream_t stream) {
  const float* x    = (const float*)d_in[0];   // (1,T,D)
  const float* emb  = (const float*)d_in[1];   // (V,E)
  const float* Wih  = (const float*)d_in[2];   // (D,3H)
  const float* Whh  = (const float*)d_in[3];   // (H,3H)
  const float* bias = (const float*)d_in[4];   // (3H)
  const float* aWih = (const float*)d_in[5];   // (D,H)
  const float* aWhh = (const float*)d_in[6];   // (H,H)
  const float* ab   = (const float*)d_in[7];   // (H)
  const float* wWih = (const float*)d_in[8];   // (E,3H)
  const float* wWhh = (const float*)d_in[9];   // (H,3H)
  const float* wb   = (const float*)d_in[10];  // (3H)
  const int* prod_word = (const int*)d_in[11]; // (T,KP)
  const int* prod_dest = (const int*)d_in[12];
  const int* prod_mask = (const int*)d_in[13];
  const int* cons_idx  = (const int*)d_in[14];
  const int* cons_mask = (const int*)d_in[15];

  float* ws     = (float*)d_ws;
  float* Gx     = ws;                              // T*3H      = 786432
  float* Ax     = Gx + (long)T_LEN * G3H;          // T*H       = 262144
  float* Wg     = Ax + (long)T_LEN * HDIM;         // T*KP*3H   = 1572864
  float* storeG = Wg + (long)T_LEN * KPRD * G3H;   // (M+1)*H   = 174592

  // store0 = zeros (graph-capture-safe)
  hipMemsetAsync(storeG, 0, (size_t)(NMATCH + 1) * HDIM * sizeof(float), stream);

  // Phase 1: time-parallel projections on the WMMA pipe (exact fp32).
  auto nblk = [](int tiles) { return (tiles + 7) / 8; };
  wmma_gemm_f32<<<nblk((T_LEN/16) * (G3H/16)), 256, 0, stream>>>(
      x, Wih, Gx, T_LEN, G3H, D_IN, nullptr);
  wmma_gemm_f32<<<nblk((T_LEN/16) * (HDIM/16)), 256, 0, stream>>>(
      x, aWih, Ax, T_LEN, HDIM, D_IN, nullptr);
  wmma_gemm_f32<<<nblk((T_LEN*KPRD/16) * (G3H/16)), 256, 0, stream>>>(
      emb, wWih, Wg, T_LEN * KPRD, G3H, EDIM, prod_word);

  // Phase 2: strict sequential chain on one persistent WGP.
  lattice_recurrence<<<1, 1024, 0, stream>>>(
      Gx, Ax, Wg, Whh, bias, aWhh, ab, wWhh, wb,
      prod_dest, prod_mask, cons_idx, cons_mask, storeG, (float*)d_out);
}